// S4_GO_BRR_45973329936665
// MI455X (gfx1250) — compile-verified
//
#include <hip/hip_runtime.h>
#include <math.h>

// Problem constants (from reference)
#define Hdim   256
#define Ldim   4096
#define Ndim   64
#define BATCH  8
#define PB     16      // batch padded to one WMMA tile of rows
#define OUTC   512     // 2*H output channels of W_out
#define CHUNKS 64      // L split for the conv reduction
#define CLEN   (Ldim / CHUNKS)   // 64 timesteps per chunk

typedef __attribute__((ext_vector_type(2))) float v2f;
typedef __attribute__((ext_vector_type(8))) float v8f;

// ---------------------------------------------------------------------------
// Kernel 0: per-(h,n) constants.
//   dtA = dt*A,  coef = C0 * B * (exp(dtA)-1)/A        (all complex, f32)
// ---------------------------------------------------------------------------
__global__ void s4_precomp_kernel(const float* __restrict__ log_dt,
                                  const float* __restrict__ A_log_re,
                                  const float* __restrict__ A_im,
                                  const float* __restrict__ B_re,
                                  const float* __restrict__ B_im,
                                  const float* __restrict__ C_re,
                                  const float* __restrict__ C_im,
                                  float* __restrict__ dtA_re,
                                  float* __restrict__ dtA_im,
                                  float* __restrict__ coef_re,
                                  float* __restrict__ coef_im) {
  int idx = blockIdx.x * blockDim.x + threadIdx.x;
  if (idx >= Hdim * Ndim) return;
  int h = idx >> 6;

  float dt  = expf(log_dt[h]);
  float Are = -expf(A_log_re[idx]);
  float Aim = A_im[idx];
  float dre = dt * Are;
  float dim = dt * Aim;

  // dA - 1
  float er = expf(dre);
  float s, c;
  sincosf(dim, &s, &c);
  float numre = er * c - 1.0f;
  float numim = er * s;

  // (dA-1)/A  via conj(A)/|A|^2
  float inv = 1.0f / (Are * Are + Aim * Aim);
  float qre = (numre * Are + numim * Aim) * inv;
  float qim = (numim * Are - numre * Aim) * inv;

  // dB = (B_re + i B_im) * q
  float bre = B_re[idx], bim = B_im[idx];
  float dBre = bre * qre - bim * qim;
  float dBim = bre * qim + bim * qre;

  // coef = C(channel 0) * dB
  float cre = C_re[idx], cim = C_im[idx];  // channel 0 = first H*N elems
  coef_re[idx] = cre * dBre - cim * dBim;
  coef_im[idx] = cre * dBim + cim * dBre;
  dtA_re[idx]  = dre;
  dtA_im[idx]  = dim;
}

// ---------------------------------------------------------------------------
// Kernel 1: k0T[l][h] = 2 * Re( sum_n coef[h,n] * exp(dtA[h,n]*l) )
// Stored transposed (l-major) so the conv kernel reads it coalesced over h.
// ---------------------------------------------------------------------------
__global__ void __launch_bounds__(Hdim)
s4_kernelgen_kernel(const float* __restrict__ dtA_re,
                    const float* __restrict__ dtA_im,
                    const float* __restrict__ coef_re,
                    const float* __restrict__ coef_im,
                    float* __restrict__ k0T) {
  int l = blockIdx.x;
  int h = threadIdx.x;
  float fl = (float)l;
  float acc = 0.0f;
  int base = h * Ndim;
#pragma unroll 4
  for (int n = 0; n < Ndim; ++n) {
    int i = base + n;
    float lr = dtA_re[i] * fl;
    float li = dtA_im[i] * fl;
    float s, c;
    sincosf(li, &s, &c);
    float e = expf(lr);
    acc += e * (coef_re[i] * c - coef_im[i] * s);
  }
  k0T[l * Hdim + h] = 2.0f * acc;
}

// ---------------------------------------------------------------------------
// Kernel 2a: partial causal conv over an L-chunk (split for occupancy).
//   part[b][chunk][h] = sum_{l in chunk} emb[ids[b,l]][h] * k0T[L-1-l][h]
// Grid (CHUNKS, BATCH) x 256 threads -> 512 blocks, no atomics.
// ---------------------------------------------------------------------------
__global__ void __launch_bounds__(Hdim)
s4_conv_partial_kernel(const int* __restrict__ ids,
                       const float* __restrict__ emb,
                       const float* __restrict__ k0T,
                       float* __restrict__ part) {
  int chunk = blockIdx.x;
  int b     = blockIdx.y;
  int h     = threadIdx.x;
  int l0    = chunk * CLEN;

  const int* idrow = ids + b * Ldim + l0;
  const float* krow = k0T + (Ldim - 1 - l0) * Hdim + h;

  float acc = 0.0f;
#pragma unroll 4
  for (int l = 0; l < CLEN; ++l) {
    int id   = idrow[l];                 // uniform across the wave
    float uv = emb[id * Hdim + h];       // coalesced over h
    float kv = krow[-l * Hdim];          // coalesced over h
    acc = fmaf(uv, kv, acc);
  }
  part[(b * CHUNKS + chunk) * Hdim + h] = acc;
}

// ---------------------------------------------------------------------------
// Kernel 2b: reduce chunks + D skip + exact GeLU into padded activation G.
// Rows b>=8 are zero padding for the 16-row WMMA tile.
// ---------------------------------------------------------------------------
__global__ void __launch_bounds__(Hdim)
s4_conv_reduce_gelu_kernel(const int* __restrict__ ids,
                           const float* __restrict__ emb,
                           const float* __restrict__ Dv,
                           const float* __restrict__ part,
                           float* __restrict__ g) {
  int b = blockIdx.x;
  int h = threadIdx.x;
  float out = 0.0f;
  if (b < BATCH) {
    float acc = 0.0f;
    const float* prow = part + b * CHUNKS * Hdim + h;
#pragma unroll 8
    for (int c = 0; c < CHUNKS; ++c) acc += prow[c * Hdim];
    int idl = ids[b * Ldim + Ldim - 1];
    float y = fmaf(emb[idl * Hdim + h], Dv[h], acc);
    out = 0.5f * y * (1.0f + erff(y * 0.70710678118654752f));
  }
  g[b * Hdim + h] = out;
}

// ---------------------------------------------------------------------------
// Kernel 3: Z(16x512) = G(16x256) x W_out^T(256x512) + b_out,
// one wave per 16x16 output tile via V_WMMA_F32_16X16X4_F32, K-loop of 64.
// A frag (16x4):  lanes 0-15 -> M=lane,K={0,1}; lanes 16-31 -> K={2,3}
// B frag (4x16):  lanes 0-15 -> N=lane,K={0,1}; lanes 16-31 -> K={2,3}
// ---------------------------------------------------------------------------
__global__ void __launch_bounds__(32)
s4_out_gemm_wmma_kernel(const float* __restrict__ g,
                        const float* __restrict__ W,     // (512,256) row-major
                        const float* __restrict__ bout,  // (512)
                        float* __restrict__ z) {         // (16,512)
  int lane = threadIdx.x;
  int n0   = blockIdx.x * 16;
  int m    = lane & 15;            // A row / B column within tile
  int kh   = (lane >> 4) << 1;     // K sub-offset: 0 or 2

  const float* arow = g + m * Hdim;
  const float* brow = W + (n0 + m) * Hdim;  // B[k][n] = W_out[n][k]

  v8f c = {};
  for (int k = 0; k < Hdim; k += 4) {
    v2f a  = *(const v2f*)(arow + k + kh);
    v2f bb = *(const v2f*)(brow + k + kh);
    c = __builtin_amdgcn_wmma_f32_16x16x4_f32(
        /*neg_a=*/false, a, /*neg_b=*/false, bb,
        /*c_mod=*/(short)0, c, /*reuse_a=*/false, /*reuse_b=*/false);
  }

  // C/D layout: VGPR v -> M=v (lanes 0-15) / M=v+8 (lanes 16-31), N=lane&15
  int rbase = (lane >> 4) << 3;
  int col   = n0 + m;
  float bv  = bout[col];
#pragma unroll
  for (int v = 0; v < 8; ++v) {
    z[(v + rbase) * OUTC + col] = c[v] + bv;
  }
}

// ---------------------------------------------------------------------------
// Kernel 4: GLU  out[b,h] = z[b,h] * sigmoid(z[b,h+256])
// ---------------------------------------------------------------------------
__global__ void s4_glu_kernel(const float* __restrict__ z,
                              float* __restrict__ out) {
  int i = blockIdx.x * blockDim.x + threadIdx.x;
  if (i >= BATCH * Hdim) return;
  int b = i >> 8;
  int h = i & 255;
  float z0 = z[b * OUTC + h];
  float z1 = z[b * OUTC + Hdim + h];
  out[i] = z0 * (1.0f / (1.0f + expf(-z1)));
}

// ---------------------------------------------------------------------------
extern "C" void kernel_launch(void* const* d_in, const int* in_sizes, int n_in,
                              void* d_out, int out_size, void* d_ws, size_t ws_size,
                              hipStream_t stream) {
  const int*   input_ids = (const int*)  d_in[0];   // (8, 4096)
  const float* embedding = (const float*)d_in[1];   // (32000, 256)
  const float* log_dt    = (const float*)d_in[2];   // (256)
  const float* A_log_re  = (const float*)d_in[3];   // (256, 64)
  const float* A_im      = (const float*)d_in[4];   // (256, 64)
  const float* B_re      = (const float*)d_in[5];   // (256, 64)
  const float* B_im      = (const float*)d_in[6];   // (256, 64)
  const float* C_re      = (const float*)d_in[7];   // (2, 256, 64)
  const float* C_im      = (const float*)d_in[8];   // (2, 256, 64)
  const float* Dv        = (const float*)d_in[9];   // (1, 256)
  const float* W_out     = (const float*)d_in[10];  // (512, 256)
  const float* b_out     = (const float*)d_in[11];  // (512)
  float*       out       = (float*)d_out;           // (8, 256)

  // Workspace carve-up (~4.8 MB)
  float* k0T     = (float*)d_ws;            // L*H = 1,048,576 floats (4 MB)
  float* dtA_re  = k0T    + Ldim * Hdim;    // H*N
  float* dtA_im  = dtA_re + Hdim * Ndim;
  float* coef_re = dtA_im + Hdim * Ndim;
  float* coef_im = coef_re + Hdim * Ndim;
  float* g       = coef_im + Hdim * Ndim;   // 16*256
  float* z       = g + PB * Hdim;           // 16*512
  float* part    = z + PB * OUTC;           // BATCH*CHUNKS*H = 131,072 floats

  s4_precomp_kernel<<<(Hdim * Ndim + 255) / 256, 256, 0, stream>>>(
      log_dt, A_log_re, A_im, B_re, B_im, C_re, C_im,
      dtA_re, dtA_im, coef_re, coef_im);

  s4_kernelgen_kernel<<<Ldim, Hdim, 0, stream>>>(
      dtA_re, dtA_im, coef_re, coef_im, k0T);

  s4_conv_partial_kernel<<<dim3(CHUNKS, BATCH), Hdim, 0, stream>>>(
      input_ids, embedding, k0T, part);

  s4_conv_reduce_gelu_kernel<<<PB, Hdim, 0, stream>>>(
      input_ids, embedding, Dv, part, g);

  s4_out_gemm_wmma_kernel<<<OUTC / 16, 32, 0, stream>>>(
      g, W_out, b_out, z);

  s4_glu_kernel<<<(BATCH * Hdim + 255) / 256, 256, 0, stream>>>(z, out);
}